// PL_64166811402730
// MI455X (gfx1250) — compile-verified
//
#include <hip/hip_runtime.h>
#include <hip/hip_bf16.h>
#include <stddef.h>

// ---------------------------------------------------------------------------
// Problem sizes (fixed by the reference)
// ---------------------------------------------------------------------------
#define B_ROWS 16384
#define D_DIM  2048
#define C_DIM  1000
#define C_PAD  1024

typedef float v2f __attribute__((ext_vector_type(2)));
typedef float v8f __attribute__((ext_vector_type(8)));

// ---------------------------------------------------------------------------
// Kernel 0: zero d_out (loss + class_freq) and colsum scratch
// ---------------------------------------------------------------------------
__global__ __launch_bounds__(256) void k_zero(float* __restrict__ out,
                                              float* __restrict__ colsum) {
  int i = blockIdx.x * 256 + threadIdx.x;
  if (i < 1001) out[i] = 0.0f;
  if (i < C_PAD) colsum[i] = 0.0f;
}

// ---------------------------------------------------------------------------
// Kernel P: pack W [2048 x 1000] into WMMA-native layout
//   Wp[(k/4)*4096 + c*4 + j] = W[k - k%4 + j][c]   (c >= 1000 -> 0)
// One thread per (k4, c) pair; float4 stores are perfectly coalesced.
// ---------------------------------------------------------------------------
__global__ __launch_bounds__(256) void k_pack_w(const float* __restrict__ W,
                                                float* __restrict__ Wp) {
  const int idx = blockIdx.x * 256 + threadIdx.x;  // 0 .. 512*1024-1
  const int k4 = idx >> 10;                        // 0..511
  const int c  = idx & (C_PAD - 1);                // 0..1023
  float4 v = make_float4(0.0f, 0.0f, 0.0f, 0.0f);
  if (c < C_DIM) {
    const float* p = W + (size_t)(k4 * 4) * C_DIM + c;
    v.x = p[0];
    v.y = p[C_DIM];
    v.z = p[2 * C_DIM];
    v.w = p[3 * C_DIM];
  }
  *(float4*)&Wp[(size_t)idx * 4] = v;
}

// ---------------------------------------------------------------------------
// Kernel 1: per-row softmax stats of y + per-class prob sums.
// ---------------------------------------------------------------------------
__global__ __launch_bounds__(256) void k_row_stats(const float* __restrict__ y,
                                                   float* __restrict__ rowmax,
                                                   float* __restrict__ rowsum,
                                                   int* __restrict__ pred,
                                                   float* __restrict__ colsum) {
  __shared__ float scol[C_PAD];
  const int tid  = threadIdx.x;
  const int wave = tid >> 5;
  const int lane = tid & 31;

  for (int c = tid; c < C_PAD; c += 256) scol[c] = 0.0f;
  __syncthreads();

  const int b = blockIdx.x * 8 + wave;
  const float* yr = y + (size_t)b * C_DIM;

  float mx = -3.4e38f;
  int mi = 0;
  for (int c = lane; c < C_DIM; c += 32) {
    float v = yr[c];
    if (v > mx) { mx = v; mi = c; }
  }
#pragma unroll
  for (int off = 1; off < 32; off <<= 1) {
    float om = __shfl_xor(mx, off, 32);
    int   oi = __shfl_xor(mi, off, 32);
    if (om > mx || (om == mx && oi < mi)) { mx = om; mi = oi; }
  }

  float se = 0.0f;
  for (int c = lane; c < C_DIM; c += 32) se += __expf(yr[c] - mx);
#pragma unroll
  for (int off = 1; off < 32; off <<= 1) se += __shfl_xor(se, off, 32);

  if (lane == 0) {
    rowmax[b] = mx;
    rowsum[b] = se;
    pred[b]   = mi;
  }

  const float irs = 1.0f / se;
  for (int c = lane; c < C_DIM; c += 32) {
    atomicAdd(&scol[c], __expf(yr[c] - mx) * irs);  // ds_add_f32
  }
  __syncthreads();
  for (int c = tid; c < C_DIM; c += 256) atomicAdd(&colsum[c], scol[c]);
}

// ---------------------------------------------------------------------------
// Kernel 2: threshold EMA update + min-max rescale into [0.8, 0.95].
// ---------------------------------------------------------------------------
__global__ __launch_bounds__(1024) void k_thresh(const float* __restrict__ th_in,
                                                 const float* __restrict__ colsum,
                                                 float* __restrict__ th_out) {
  __shared__ float smn[1024];
  __shared__ float smx[1024];
  const int c = threadIdx.x;

  float t  = 0.0f;
  float mn = 3.4e38f, mx = -3.4e38f;
  if (c < C_DIM) {
    t = th_in[c] * 0.99f + 0.01f * (colsum[c] * (1.0f / (float)B_ROWS));
    mn = t; mx = t;
  }
  smn[c] = mn;
  smx[c] = mx;
  __syncthreads();
#pragma unroll
  for (int s = 512; s > 0; s >>= 1) {
    if (c < s) {
      smn[c] = fminf(smn[c], smn[c + s]);
      smx[c] = fmaxf(smx[c], smx[c + s]);
    }
    __syncthreads();
  }
  const float lo = smn[0], hi = smx[0];
  const float v = (c < C_DIM)
                      ? ((t - lo) / (hi - lo) * 0.15f + 0.8f)
                      : 3.0e38f;
  th_out[c] = v;
}

// ---------------------------------------------------------------------------
// Kernel 3: fused f32-WMMA GEMM (x@W) + log-softmax + p_target dot + loss
//           + gt-threshold histogram.
// Block: 256 threads (8 wave32s) -> 32 rows x all 1024 (padded) columns.
// Each wave: 2 row groups x 8 column tiles = 16 accumulators; branch-free,
// software-pipelined inner loop over the packed Wp layout.
// ---------------------------------------------------------------------------
__global__ __launch_bounds__(256) void k_fused_gemm_loss(
    const float* __restrict__ x, const float* __restrict__ y,
    const float* __restrict__ mask, const float* __restrict__ Wp,
    const float* __restrict__ rowmax, const float* __restrict__ rowsum,
    const int* __restrict__ pred, const float* __restrict__ th,
    float* __restrict__ out) {
  __shared__ float xs[32 * 136];  // 32 rows x 128 cols, padded stride
  __shared__ float redf[32][8];
  __shared__ float redf2[32][8];
  __shared__ int   redi[32][8];
  __shared__ float s_rm[32], s_irs[32], s_msk[32];
  __shared__ int   s_pred[32];
  __shared__ float s_M[32], s_ML[32];
  __shared__ int   s_gt[32];

  const int tid  = threadIdx.x;
  const int wave = tid >> 5;   // 0..7
  const int lane = tid & 31;
  const int half = lane >> 4;  // 0 / 1
  const int l16  = lane & 15;
  const int m0   = blockIdx.x * 32;

  if (tid < 32) {
    const int b = m0 + tid;
    s_rm[tid]   = rowmax[b];
    s_irs[tid]  = 1.0f / rowsum[b];
    s_pred[tid] = pred[b];
    s_msk[tid]  = mask[b];
  }

  v8f acc[2][8];
#pragma unroll
  for (int g = 0; g < 2; ++g)
#pragma unroll
    for (int t = 0; t < 8; ++t)
#pragma unroll
      for (int i = 0; i < 8; ++i) acc[g][t][i] = 0.0f;

  // Per-lane float offset into a 4096-float Wp "k4 row"
  const int bofs = (wave * 128 + l16) * 4 + half * 2;

  // ------------------------- K loop: 2048 in chunks of 128 ------------------
  for (int k0 = 0; k0 < D_DIM; k0 += 128) {
    __syncthreads();  // protect xs from previous iteration's readers
#pragma unroll
    for (int it = 0; it < 4; ++it) {
      const int v  = tid + it * 256;  // 0..1023 float4 slots (32 rows x 32)
      const int r  = v >> 5;
      const int c4 = (v & 31) << 2;
      const float4 src =
          *(const float4*)&x[(size_t)(m0 + r) * D_DIM + k0 + c4];
      *(float4*)&xs[r * 136 + c4] = src;
    }
    __syncthreads();

    // Prologue of the pipelined inner loop
    const float* wp = Wp + (size_t)(k0 >> 2) * 4096 + bofs;
    v2f a0 = *(const v2f*)&xs[l16 * 136 + half * 2];
    v2f a1 = *(const v2f*)&xs[(16 + l16) * 136 + half * 2];
    v2f bf[8];
#pragma unroll
    for (int t = 0; t < 8; ++t) bf[t] = *(const v2f*)(wp + t * 64);

#pragma unroll 2
    for (int k4 = 0; k4 < 128; k4 += 4) {
      // Prefetch next step (A reads land in xs pad region on last step;
      // B reads land in the Wp slack row on the very last step — both safe,
      // values dead after loop exit).
      const v2f a0n = *(const v2f*)&xs[l16 * 136 + (k4 + 4) + half * 2];
      const v2f a1n = *(const v2f*)&xs[(16 + l16) * 136 + (k4 + 4) + half * 2];
      const float* wpn = wp + 4096;
      v2f bfn[8];
#pragma unroll
      for (int t = 0; t < 8; ++t) bfn[t] = *(const v2f*)(wpn + t * 64);

#pragma unroll
      for (int t = 0; t < 8; ++t)
        acc[0][t] = __builtin_amdgcn_wmma_f32_16x16x4_f32(
            false, a0, false, bf[t], (short)0, acc[0][t], false, false);
#pragma unroll
      for (int t = 0; t < 8; ++t)
        acc[1][t] = __builtin_amdgcn_wmma_f32_16x16x4_f32(
            false, a1, false, bf[t], (short)0, acc[1][t], false, false);

      a0 = a0n;
      a1 = a1n;
#pragma unroll
      for (int t = 0; t < 8; ++t) bf[t] = bfn[t];
      wp = wpn;
    }
  }

  // ------------------------------ Epilogue ----------------------------------
  float thv[8];
#pragma unroll
  for (int t = 0; t < 8; ++t) {
    const int ct = (wave * 8 + t) * 16 + l16;
    thv[t] = (ct < C_DIM) ? th[ct] : 3.0e38f;
  }

  // Round A: per-row max(out) + first gt index from y_probs
#pragma unroll
  for (int g = 0; g < 2; ++g) {
#pragma unroll
    for (int r = 0; r < 8; ++r) {
      const int row = g * 16 + half * 8 + r;
      const float rm = s_rm[row], irs = s_irs[row];
      float mx = -3.4e38f;
      int cand = 0x7fffffff;
#pragma unroll
      for (int t = 0; t < 8; ++t) {
        const int ct = (wave * 8 + t) * 16 + l16;
        if (ct < C_DIM) {
          mx = fmaxf(mx, acc[g][t][r]);
          const float yp =
              __expf(y[(size_t)(m0 + row) * C_DIM + ct] - rm) * irs;
          if (yp > thv[t]) cand = min(cand, ct);
        }
      }
#pragma unroll
      for (int off = 1; off < 16; off <<= 1) {
        mx   = fmaxf(mx, __shfl_xor(mx, off, 32));
        cand = min(cand, __shfl_xor(cand, off, 32));
      }
      if (l16 == 0) { redf[row][wave] = mx; redi[row][wave] = cand; }
    }
  }
  __syncthreads();
  if (tid < 32) {
    float mx = redf[tid][0];
    int cand = redi[tid][0];
#pragma unroll
    for (int wv = 1; wv < 8; ++wv) {
      mx   = fmaxf(mx, redf[tid][wv]);
      cand = min(cand, redi[tid][wv]);
    }
    s_M[tid] = mx;
    const int ga = (cand != 0x7fffffff) ? 1 : 0;
    s_gt[tid] = ga;
    atomicAdd(&out[1 + (ga ? cand : 0)], 1.0f);  // class_freq histogram
  }
  __syncthreads();

  // Round B: log-sum-exp of out
#pragma unroll
  for (int g = 0; g < 2; ++g) {
#pragma unroll
    for (int r = 0; r < 8; ++r) {
      const int row = g * 16 + half * 8 + r;
      const float M = s_M[row];
      float se = 0.0f;
#pragma unroll
      for (int t = 0; t < 8; ++t) {
        const int ct = (wave * 8 + t) * 16 + l16;
        if (ct < C_DIM) se += __expf(acc[g][t][r] - M);
      }
#pragma unroll
      for (int off = 1; off < 16; off <<= 1) se += __shfl_xor(se, off, 32);
      if (l16 == 0) redf[row][wave] = se;
    }
  }
  __syncthreads();
  if (tid < 32) {
    float se = 0.0f;
#pragma unroll
    for (int wv = 0; wv < 8; ++wv) se += redf[tid][wv];
    s_ML[tid] = s_M[tid] + __logf(se);
  }
  __syncthreads();

  // Round C: dot(p_target, out) and sum(p_target)
#pragma unroll
  for (int g = 0; g < 2; ++g) {
#pragma unroll
    for (int r = 0; r < 8; ++r) {
      const int row = g * 16 + half * 8 + r;
      const float rm = s_rm[row], irs = s_irs[row];
      const int gm = s_gt[row];
      const int pr = s_pred[row];
      float dp = 0.0f, wsum = 0.0f;
#pragma unroll
      for (int t = 0; t < 8; ++t) {
        const int ct = (wave * 8 + t) * 16 + l16;
        if (ct < C_DIM) {
          float w;
          if (gm) {
            w = (ct == pr) ? 10.0f : 0.0f;
          } else {
            w = __expf(y[(size_t)(m0 + row) * C_DIM + ct] - rm) * irs;
          }
          dp   += w * acc[g][t][r];
          wsum += w;
        }
      }
#pragma unroll
      for (int off = 1; off < 16; off <<= 1) {
        dp   += __shfl_xor(dp, off, 32);
        wsum += __shfl_xor(wsum, off, 32);
      }
      if (l16 == 0) { redf[row][wave] = dp; redf2[row][wave] = wsum; }
    }
  }
  __syncthreads();
  if (tid < 32) {
    float dp = 0.0f, wsum = 0.0f;
#pragma unroll
    for (int wv = 0; wv < 8; ++wv) {
      dp   += redf[tid][wv];
      wsum += redf2[tid][wv];
    }
    // sum(p * log_softmax) = dot(p,out) - sum(p)*(M + log(sumexp))
    const float lossb =
        -(dp - wsum * s_ML[tid]) * s_msk[tid] * (1.0f / (float)B_ROWS);
    atomicAdd(&out[0], lossb);
  }
}

// ---------------------------------------------------------------------------
// Launch
// ---------------------------------------------------------------------------
extern "C" void kernel_launch(void* const* d_in, const int* in_sizes, int n_in,
                              void* d_out, int out_size, void* d_ws,
                              size_t ws_size, hipStream_t stream) {
  const float* x    = (const float*)d_in[0];  // [16384, 2048]
  const float* y    = (const float*)d_in[1];  // [16384, 1000]
  const float* mask = (const float*)d_in[2];  // [16384]
  const float* thp  = (const float*)d_in[3];  // [1000]
  const float* W    = (const float*)d_in[4];  // [2048, 1000]
  float* out = (float*)d_out;                 // [0]=loss, [1..1000]=class_freq

  // Workspace layout (floats):
  //   [0,16384)        rowmax
  //   [16384,32768)    rowsum
  //   [32768,49152)    pred (int)
  //   [49152,50176)    colsum
  //   [50176,51200)    th
  //   [51200, +512*4096 + 4096)  packed Wp (+1 slack row for prefetch)
  float* wsf    = (float*)d_ws;
  float* rowmax = wsf;
  float* rowsum = wsf + 16384;
  int*   pred   = (int*)(wsf + 32768);
  float* colsum = wsf + 49152;
  float* th     = wsf + 50176;
  float* Wp     = wsf + 51200;

  k_zero<<<5, 256, 0, stream>>>(out, colsum);
  k_pack_w<<<(512 * C_PAD) / 256, 256, 0, stream>>>(W, Wp);
  k_row_stats<<<B_ROWS / 8, 256, 0, stream>>>(y, rowmax, rowsum, pred, colsum);
  k_thresh<<<1, 1024, 0, stream>>>(thp, colsum, th);
  k_fused_gemm_loss<<<B_ROWS / 32, 256, 0, stream>>>(
      x, y, mask, Wp, rowmax, rowsum, pred, th, out);
}